// TransducerLoss_6554120094303
// MI455X (gfx1250) — compile-verified
//
#include <hip/hip_runtime.h>
#include <hip/hip_bf16.h>
#include <cstdint>

#define NEGINF (-1e30f)

#ifndef __has_builtin
#define __has_builtin(x) 0
#endif

#if __has_builtin(__builtin_amdgcn_global_load_async_to_lds_b128) && \
    __has_builtin(__builtin_amdgcn_s_wait_asynccnt)
#define USE_ASYNC_LDS 1
#else
#define USE_ASYNC_LDS 0
#endif

typedef int v4i_async __attribute__((ext_vector_type(4)));
#define AS1CAST(p) ((__attribute__((address_space(1))) v4i_async*)(uintptr_t)(p))
#define AS3CAST(p) ((__attribute__((address_space(3))) v4i_async*)(uintptr_t)(p))

// ---------------------------------------------------------------------------
// Kernel 1: fused logsumexp over V + gather of blank/emit log-probs.
// One wave32 per (b,t,u) row of V floats; 8 waves per block.
// Row is staged global -> LDS via CDNA5 async-to-LDS (fallback: plain loads),
// reduced with wave32 shuffles.
// ---------------------------------------------------------------------------
__global__ __launch_bounds__(256) void k_lse(const float* __restrict__ logits,
                                             const int* __restrict__ labels,
                                             float* __restrict__ blank_ws,
                                             float* __restrict__ emit_ws,
                                             int B, int T, int U, int V, int rows)
{
    __shared__ float tile[8][512];          // 16 KB: one 512-float row per wave
    const int wave = threadIdx.x >> 5;
    const int lane = threadIdx.x & 31;
    const int row  = blockIdx.x * 8 + wave;
    if (row >= rows) return;                // uniform per wave; no block barriers used

    const int U1 = U + 1;
    const float* rowp = logits + (size_t)row * V;
    float* lds = tile[wave];

    // ---- stage row into LDS (async data mover path on CDNA5) ----
#if USE_ASYNC_LDS
    for (int e = lane * 4; e + 3 < V; e += 128) {
        __builtin_amdgcn_global_load_async_to_lds_b128(
            AS1CAST(rowp + e), AS3CAST(lds + e), 0, 0);
    }
    __builtin_amdgcn_s_wait_asynccnt(0);
    __asm__ volatile("" ::: "memory");
#else
    for (int e = lane * 4; e + 3 < V; e += 128) {
        const float4 t4 = *reinterpret_cast<const float4*>(rowp + e);
        *reinterpret_cast<float4*>(lds + e) = t4;
    }
#endif

    // ---- pass 1: row max (wave32 shuffle reduction) ----
    float m = -3.0e38f;
    for (int e = lane * 4; e + 3 < V; e += 128) {
        const float4 x = *reinterpret_cast<const float4*>(lds + e);
        m = fmaxf(m, fmaxf(fmaxf(x.x, x.y), fmaxf(x.z, x.w)));
    }
    for (int s = 16; s > 0; s >>= 1) m = fmaxf(m, __shfl_xor(m, s, 32));

    // ---- pass 2: sum exp(x - m) ----
    float sum = 0.f;
    for (int e = lane * 4; e + 3 < V; e += 128) {
        const float4 x = *reinterpret_cast<const float4*>(lds + e);
        sum += expf(x.x - m) + expf(x.y - m) + expf(x.z - m) + expf(x.w - m);
    }
    for (int s = 16; s > 0; s >>= 1) sum += __shfl_xor(sum, s, 32);
    const float lse = m + logf(sum);

    if (lane == 0) {
        const int u  = row % U1;
        const int bt = row / U1;
        const int t  = bt % T;
        const int b  = bt / T;
        blank_ws[row] = lds[0] - lse;                       // lp[t,u,BLANK]
        if (u < U) {
            int lab = labels[b * U + u];
            lab = lab < 0 ? 0 : (lab > V - 1 ? V - 1 : lab);
            emit_ws[((size_t)b * T + t) * U + u] = lds[lab] - lse;
        }
    }
}

// ---------------------------------------------------------------------------
// Kernel 2: RNN-T forward DP, anti-diagonal wavefront. One block per sample.
// alpha[t][u] = logaddexp(alpha[t][u-1] + emit[t][u-1],
//                         alpha[t-1][u] + blank[t-1][u]),  alpha[0][0] = 0.
// nll[b] = -(alpha[tlen-1][ulen] + blank[tlen-1][ulen]).
// ---------------------------------------------------------------------------
__global__ __launch_bounds__(256) void k_dp(const float* __restrict__ blank_ws,
                                            const float* __restrict__ emit_ws,
                                            const int* __restrict__ label_lengths,
                                            const int* __restrict__ attn,
                                            float* __restrict__ nll,
                                            int T, int U)
{
    const int b  = blockIdx.x;
    const int U1 = U + 1;
    __shared__ float bufA[1032];
    __shared__ float bufB[1032];
    __shared__ int   s_sum;
    const int tid = threadIdx.x;

    if (tid == 0) s_sum = 0;
    __syncthreads();
    int part = 0;
    for (int i = tid; i < T; i += blockDim.x) part += attn[b * T + i];
    if (part) atomicAdd(&s_sum, part);
    __syncthreads();
    int tlen = s_sum;              tlen = tlen < 1 ? 1 : (tlen > T ? T : tlen);
    int ulen = label_lengths[b];   ulen = ulen < 0 ? 0 : (ulen > U ? U : ulen);

    const float* Bk = blank_ws + (size_t)b * T * U1;   // blank[t][u] = Bk[t*U1+u]
    const float* Em = emit_ws  + (size_t)b * T * U;    // emit[t][u]  = Em[t*U+u]

    float* prev = bufA;
    float* cur  = bufB;

    for (int d = 0; d <= T - 1 + U; ++d) {
        const int u = tid;
        const int t = d - u;
        const bool active = (u <= U) && (t >= 0) && (t < T);
        float val = NEGINF;
        if (active) {
            if (d == 0) {
                val = 0.f;
            } else {
                const float left = (u > 0) ? prev[u - 1] + Em[(size_t)t * U + (u - 1)] : NEGINF;
                const float up   = (t > 0) ? prev[u]     + Bk[(size_t)(t - 1) * U1 + u] : NEGINF;
                const float mx = fmaxf(left, up);
                const float mn = fminf(left, up);
                val = (mx <= -1e29f) ? NEGINF : mx + log1pf(expf(mn - mx));
            }
        }
        if (u <= U) cur[u] = val;
        if (active && t == tlen - 1 && u == ulen)
            nll[b] = -(val + Bk[(size_t)t * U1 + u]);
        __syncthreads();
        float* tmp = prev; prev = cur; cur = tmp;
    }
}

// ---------------------------------------------------------------------------
// Kernel 3: mean over batch -> scalar output.
// ---------------------------------------------------------------------------
__global__ void k_mean(const float* __restrict__ nll, float* __restrict__ out, int B)
{
    if (threadIdx.x == 0 && blockIdx.x == 0) {
        float s = 0.f;
        for (int i = 0; i < B; ++i) s += nll[i];
        out[0] = s / (float)B;
    }
}

extern "C" void kernel_launch(void* const* d_in, const int* in_sizes, int n_in,
                              void* d_out, int out_size, void* d_ws, size_t ws_size,
                              hipStream_t stream)
{
    const float* logits        = (const float*)d_in[0];
    const int*   labels        = (const int*)d_in[1];
    const int*   label_lengths = (const int*)d_in[2];
    const int*   attn          = (const int*)d_in[3];

    const int B  = in_sizes[2];
    const int U  = in_sizes[1] / B;
    const int T  = in_sizes[3] / B;
    const int U1 = U + 1;
    const long long rows = (long long)B * T * U1;
    const int V  = (int)((long long)in_sizes[0] / rows);
    const int R  = (int)rows;

    float* ws       = (float*)d_ws;
    float* blank_ws = ws;                               // R floats
    float* emit_ws  = blank_ws + rows;                  // B*T*U floats
    float* nll      = emit_ws + (long long)B * T * U;   // B floats

    k_lse<<<dim3((R + 7) / 8), dim3(256), 0, stream>>>(
        logits, labels, blank_ws, emit_ws, B, T, U, V, R);
    k_dp<<<dim3(B), dim3(256), 0, stream>>>(
        blank_ws, emit_ws, label_lengths, attn, nll, T, U);
    k_mean<<<dim3(1), dim3(32), 0, stream>>>(nll, (float*)d_out, B);
}